// B2Q_Net_52166672777295
// MI455X (gfx1250) — compile-verified
//
#include <hip/hip_runtime.h>
#include <hip/hip_bf16.h>
#include <stdint.h>

// ---- problem constants (match reference) ----
#define T_DIM   16384
#define C_DIM   2048
#define NCLASS  10
#define NW      20
#define NWP1    21          // NW+1
#define HALFW   10
#define NTOKEN  64
#define NCOMB   32          // padded combined N (10 g cols + 21 l cols + 1 pad)
#define SSTRIDE 16          // padded stride for s[t,c]

// ---- GEMM tiling ----
#define ROWS_WG 128         // rows per workgroup (8 waves x 16 rows)
#define KT      32          // K columns per LDS chunk
#define NCHUNK  (C_DIM / KT)        // 64 chunks
#define LDS_ROW 34          // KT + 2 pad DWORDs (bank-conflict-free, 8B aligned)

typedef float v2f __attribute__((ext_vector_type(2)));
typedef float v8f __attribute__((ext_vector_type(8)));
typedef unsigned int u32x4 __attribute__((ext_vector_type(4)));
typedef int i32x4 __attribute__((ext_vector_type(4)));
typedef int i32x8 __attribute__((ext_vector_type(8)));

#if __has_builtin(__builtin_amdgcn_tensor_load_to_lds)
#define HAVE_TDM 1
#else
#define HAVE_TDM 0
#endif

// ------------------------------------------------------------------
// 1) Pack Wg (2048x10) + Wl (2048x21) directly into WMMA B-fragment
//    order: Bp[((ks*2 + blk)*32 + lane)*2 + j] = Wc[(4ks+2*(lane>>4)+j)*32 + 16blk + lane&15]
//    so the GEMM inner loop loads B as coalesced per-lane v2f.
// ------------------------------------------------------------------
__global__ void build_bpack_kernel(const float* __restrict__ Wg,
                                   const float* __restrict__ Wl,
                                   float* __restrict__ Bp) {
    int i = blockIdx.x * blockDim.x + threadIdx.x;   // 0 .. 512*2*32*2-1
    int j  = i & 1;
    int l  = (i >> 1) & 31;
    int b  = (i >> 6) & 1;
    int ks = i >> 7;                                 // 0..511  (K/4 steps)
    int k  = 4 * ks + 2 * (l >> 4) + j;              // K row
    int n  = 16 * b + (l & 15);                      // combined N col
    float v = 0.0f;
    if (n < NCLASS)             v = Wg[k * NCLASS + n];
    else if (n < NCLASS + NWP1) v = Wl[k * NWP1 + (n - NCLASS)];
    Bp[i] = v;
}

// ------------------------------------------------------------------
// TDM issue: 2D tile (KT x ROWS_WG) of F, fp32, into LDS with
// 2-DWORD row padding (pad_interval = 32 DWORDs -> LDS_ROW = 34 floats).
// Descriptor built per CDNA5 ISA 8.3-8.6 (6-arg toolchain form).
// ------------------------------------------------------------------
__device__ __forceinline__ void tdm_issue_tile(const float* F, int row0,
                                               int chunk, unsigned lds_off) {
#if HAVE_TDM
    unsigned long long ga =
        (unsigned long long)(uintptr_t)(F + (size_t)row0 * C_DIM + (size_t)chunk * KT);
    u32x4 g0;
    g0[0] = 1u;                                     // count=1, user descriptor
    g0[1] = lds_off;                                // lds_addr (bytes)
    g0[2] = (unsigned)ga;                           // global_addr[31:0]
    g0[3] = (unsigned)((ga >> 32) & 0x01FFFFFFu)    // global_addr[56:32]
            | 0x80000000u;                          // type=2 ("image")
    i32x8 g1;
    g1[0] = (2 << 16)     // data_size = 4 bytes
          | (1 << 20)     // pad_enable
          | (4 << 22)     // pad_interval: 2<<4 = 32 DWORDs
          | (1 << 25);    // pad_amount: 2 DWORDs
    g1[1] = (int)((C_DIM & 0xFFFF) << 16);          // tensor_dim0[15:0] @ [31:16]
    g1[2] = (int)(((C_DIM >> 16) & 0xFFFF)          // tensor_dim0[31:16]
          | ((T_DIM & 0xFFFF) << 16));              // tensor_dim1[15:0]
    g1[3] = (int)(((T_DIM >> 16) & 0xFFFF)          // tensor_dim1[31:16]
          | (KT << 16));                            // tile_dim0 = KT
    g1[4] = ROWS_WG;                                // tile_dim1 (tile_dim2 = 0)
    g1[5] = C_DIM;                                  // tensor_dim0_stride[31:0]
    g1[6] = 0;                                      // stride hi / dim1_stride (unused, 2D)
    g1[7] = 0;
    i32x4 g2 = {0, 0, 0, 0};                        // dims 2/3 unused
    i32x4 g3 = {0, 0, 0, 0};                        // dim 4 unused
    i32x8 g4 = {0, 0, 0, 0, 0, 0, 0, 0};            // extra operand (6-arg form)
    __builtin_amdgcn_tensor_load_to_lds(g0, g1, g2, g3, g4, 0);
#endif
}

// ------------------------------------------------------------------
// 2) Streaming skinny GEMM: gl[T,32] = F[T,2048] @ Wc[2048,32]
//    A tiles: TDM double-buffered into LDS; B: packed coalesced fragments;
//    one wave per 16-row tile, V_WMMA_F32_16X16X4_F32 x2 per k-step.
// ------------------------------------------------------------------
__global__ __launch_bounds__(256)
void gemm_wmma_tdm_kernel(const float* __restrict__ F,
                          const float* __restrict__ Bpack,
                          float* __restrict__ gl) {
    __shared__ float abuf[2][ROWS_WG * LDS_ROW + 4];

    const int wave = threadIdx.x >> 5;
    const int lane = threadIdx.x & 31;
    const int hw   = lane >> 4;         // half-wave: K sub-pair select
    const int l16  = lane & 15;
    const int row0 = blockIdx.x * ROWS_WG;

    v8f acc0 = {};   // N = 0..15
    v8f acc1 = {};   // N = 16..31

    // per-lane A-fragment read offset within a buffer (floats)
    const int aoff = (wave * 16 + l16) * LDS_ROW + 2 * hw;

#if HAVE_TDM
    if (wave == 0)
        tdm_issue_tile(F, row0, 0, (unsigned)(uintptr_t)&abuf[0][0]);
#endif

    for (int c = 0; c < NCHUNK; ++c) {
#if HAVE_TDM
        if (wave == 0) __builtin_amdgcn_s_wait_tensorcnt(0);
        __syncthreads();                         // chunk c visible to all waves
        if (wave == 0 && c + 1 < NCHUNK)
            tdm_issue_tile(F, row0, c + 1,
                           (unsigned)(uintptr_t)&abuf[(c + 1) & 1][0]);
        const float* ab = &abuf[c & 1][aoff];
#else
        // fallback: cooperative synchronous load of the chunk
        __syncthreads();
        for (int t = threadIdx.x; t < ROWS_WG * KT; t += 256) {
            int r = t >> 5, col = t & (KT - 1);
            abuf[0][r * LDS_ROW + col] =
                F[(size_t)(row0 + r) * C_DIM + (size_t)c * KT + col];
        }
        __syncthreads();
        const float* ab = &abuf[0][aoff];
#endif
        const float* bp = Bpack + (size_t)(c * (KT / 4)) * 128 + lane * 2;
#pragma unroll
        for (int kk = 0; kk < KT / 4; ++kk) {
            v2f a  = *(const v2f*)(ab + kk * 4);            // ds_load_b64
            v2f b0 = *(const v2f*)(bp + kk * 128);          // coalesced b64
            v2f b1 = *(const v2f*)(bp + kk * 128 + 64);
            acc0 = __builtin_amdgcn_wmma_f32_16x16x4_f32(
                       false, a, false, b0, (short)0, acc0, false, false);
            acc1 = __builtin_amdgcn_wmma_f32_16x16x4_f32(
                       false, a, false, b1, (short)0, acc1, false, false);
        }
        __syncthreads();                         // compute done before buffer reuse
    }

    // C/D layout: VGPR r -> (M = r + 8*hw, N = l16 [+16])
    float* out = gl + (size_t)(row0 + wave * 16 + 8 * hw) * NCOMB + l16;
#pragma unroll
    for (int r = 0; r < 8; ++r) {
        out[(size_t)r * NCOMB]      = acc0[r];
        out[(size_t)r * NCOMB + 16] = acc1[r];
    }
}

// ------------------------------------------------------------------
// 3) Sliding-window softmax-sum: s[t,c] = sum_w softmax_w(g[t+w-10,c]+l[t,w])
// ------------------------------------------------------------------
__global__ void softmax_sum_kernel(const float* __restrict__ gl,
                                   float* __restrict__ sbuf) {
    int i = blockIdx.x * blockDim.x + threadIdx.x;
    if (i >= T_DIM * NCLASS) return;
    int t = i / NCLASS;
    int c = i - t * NCLASS;

    float v[NWP1];
    float m = -INFINITY;
#pragma unroll
    for (int w = 0; w < NWP1; ++w) {
        int tp = t + w - HALFW;
        float gv = (tp >= 0 && tp < T_DIM) ? gl[(size_t)tp * NCOMB + c] : 0.0f;
        float lv = gl[(size_t)t * NCOMB + NCLASS + w];
        v[w] = gv + lv;
        m = fmaxf(m, v[w]);
    }
    float S = 0.0f;
#pragma unroll
    for (int w = 0; w < NWP1; ++w) S += expf(v[w] - m);
    float s = 0.0f;
#pragma unroll
    for (int w = 0; w < NWP1; ++w) {
        float sc = expf(v[w] - m) / S;
        if (!(sc == sc)) sc = 0.0f;      // NaN -> 0, as in reference
        s += sc;
    }
    sbuf[(size_t)t * SSTRIDE + c] = s;
}

// ------------------------------------------------------------------
// 4) phase[t] = max_c sum_{d=-10}^{+9} s[t+d, c]   (zero-padded)
// ------------------------------------------------------------------
__global__ void phase_kernel(const float* __restrict__ sbuf,
                             float* __restrict__ phase) {
    int t = blockIdx.x * blockDim.x + threadIdx.x;
    if (t >= T_DIM) return;
    float best = -INFINITY;
#pragma unroll
    for (int c = 0; c < NCLASS; ++c) {
        float a = 0.0f;
        for (int d = -HALFW; d < HALFW; ++d) {
            int tp = t + d;
            if (tp >= 0 && tp < T_DIM) a += sbuf[(size_t)tp * SSTRIDE + c];
        }
        best = fmaxf(best, a);
    }
    phase[t] = best;
}

// ------------------------------------------------------------------
// 5) Exact stable top-64 (ties -> lowest index, like jax.lax.top_k)
// ------------------------------------------------------------------
__global__ __launch_bounds__(1024)
void topk_kernel(const float* __restrict__ phase,
                 int* __restrict__ idx_out,
                 float* __restrict__ out_tail) {
    __shared__ unsigned long long red[1024];
    __shared__ unsigned char taken[T_DIM];     // 16 KB
    const int tid = threadIdx.x;

    for (int i = tid; i < T_DIM; i += 1024) taken[i] = 0;
    __syncthreads();

    for (int it = 0; it < NTOKEN; ++it) {
        unsigned long long best = 0ull;
        for (int i = tid; i < T_DIM; i += 1024) {
            if (taken[i]) continue;
            unsigned int b = __float_as_uint(phase[i]);
            b = (b & 0x80000000u) ? ~b : (b | 0x80000000u);   // total order
            unsigned long long key =
                ((unsigned long long)b << 32) | (unsigned int)(T_DIM - 1 - i);
            if (key > best) best = key;
        }
        red[tid] = best;
        __syncthreads();
        for (int off = 512; off > 0; off >>= 1) {
            if (tid < off && red[tid + off] > red[tid]) red[tid] = red[tid + off];
            __syncthreads();
        }
        if (tid == 0) {
            int sel = T_DIM - 1 - (int)(red[0] & 0xFFFFFFFFu);
            idx_out[it]  = sel;
            out_tail[it] = (float)sel;
            taken[sel] = 1;
        }
        __syncthreads();
    }
}

// ------------------------------------------------------------------
// 6) Gather the 64 selected rows of frame_feature into d_out
// ------------------------------------------------------------------
__global__ void gather_kernel(const float* __restrict__ F,
                              const int* __restrict__ idx,
                              float* __restrict__ out) {
    int row = blockIdx.x;
    int src = idx[row];
    const float4* s = (const float4*)(F + (size_t)src * C_DIM);
    float4*       d = (float4*)(out + (size_t)row * C_DIM);
    for (int j = threadIdx.x; j < C_DIM / 4; j += blockDim.x) d[j] = s[j];
}

// ------------------------------------------------------------------
extern "C" void kernel_launch(void* const* d_in, const int* in_sizes, int n_in,
                              void* d_out, int out_size, void* d_ws, size_t ws_size,
                              hipStream_t stream) {
    const float* F  = (const float*)d_in[0];   // (T,B=1,C) fp32
    const float* Wg = (const float*)d_in[1];   // (C,10)
    const float* Wl = (const float*)d_in[2];   // (C,21)
    float* out = (float*)d_out;                // [64*2048 gathered | 64 indices]

    // workspace layout (floats)
    float* Bp = (float*)d_ws;                        // 512*128      =  65536
    float* gl = Bp + (size_t)(C_DIM / 4) * 128;      // 16384*32     = 524288
    float* sb = gl + (size_t)T_DIM * NCOMB;          // 16384*16     = 262144
    float* ph = sb + (size_t)T_DIM * SSTRIDE;        // 16384
    int*   ix = (int*)(ph + T_DIM);                  // 64 ints

    build_bpack_kernel<<<(C_DIM * NCOMB) / 256, 256, 0, stream>>>(Wg, Wl, Bp);
    gemm_wmma_tdm_kernel<<<T_DIM / ROWS_WG, 256, 0, stream>>>(F, Bp, gl);
    softmax_sum_kernel<<<(T_DIM * NCLASS + 255) / 256, 256, 0, stream>>>(gl, sb);
    phase_kernel<<<T_DIM / 256, 256, 0, stream>>>(sb, ph);
    topk_kernel<<<1, 1024, 0, stream>>>(ph, ix, out + (size_t)NTOKEN * C_DIM);
    gather_kernel<<<NTOKEN, 256, 0, stream>>>(F, ix, out);
}